// FrameDockingScoreModel_52484500357800
// MI455X (gfx1250) — compile-verified
//
#include <hip/hip_runtime.h>
#include <math.h>

typedef float v2f __attribute__((ext_vector_type(2)));
typedef float v8f __attribute__((ext_vector_type(8)));

#define WPB   4            // waves per block
#define BLOCK (WPB * 32)

// normalization constants
#define IS3   0.5773502691896258f    // 1/sqrt(3)
#define IS2   0.7071067811865476f    // 1/sqrt(2)
#define RS21  0.21821789023599239f   // 1/sqrt(21)
#define RS10  0.31622776601683794f   // 1/sqrt(10)
#define RS105 0.09759000729485332f   // 1/sqrt(105)
#define RS50  0.14142135623730950f   // 1/sqrt(50)

// Gaussian smear coefficients pre-multiplied by log2(e) so we can use raw v_exp_f32:
//   exp(c*t^2) = exp2(c*log2e * t^2)
#define C5L2  (-0.32f * 1.4426950408889634f)   // num=5 smear
#define C4L2  (-0.18f * 1.4426950408889634f)   // num=4 smear

__launch_bounds__(BLOCK, 2)
__global__ void frame_dock_kernel(
    const float* __restrict__ lig,    const float* __restrict__ rec,
    const float* __restrict__ W1_sso, const float* __restrict__ W1_svv,
    const float* __restrict__ W1_vsv, const float* __restrict__ W1_vvo,
    const float* __restrict__ W1_vvv, const float* __restrict__ W2_sso,
    const float* __restrict__ W2_svv, const float* __restrict__ W2_vsv,
    const float* __restrict__ W2_vvo, const float* __restrict__ W2_vvv,
    float* __restrict__ out, int E)
{
    // B-fragment layout for V_WMMA_F32_16X16X4_F32 (K x N = 4 x 16, f32):
    //   lane L, vgpr v  ->  element (K = 2*(L/16) + v, N = L%16)
    __shared__ v2f   w1b[10 * 32];       // layer-1 B tiles: [nt*5+kk][lane]
    __shared__ v2f   w2b[25 * 32];       // layer-2 B tiles: [kk][lane]
    __shared__ float sc[WPB][16 * 32];   // per-wave D-tile transpose scratch

    const int tid = threadIdx.x;

    // ---- repack weights into WMMA B-fragment layout (once per block) ----
    {
        float* f1 = reinterpret_cast<float*>(w1b);
        float* f2 = reinterpret_cast<float*>(w2b);
        for (int idx = tid; idx < 640 + 1600; idx += BLOCK) {
            if (idx < 640) {
                const int v  = idx & 1;
                const int ln = (idx >> 1) & 31;
                const int kt = idx >> 6;            // 0..9 = nt*5+kk
                const int kk = kt % 5, nt = kt / 5;
                const int K  = 4 * kk + 2 * (ln >> 4) + v;   // 0..19 (padded to 20)
                const int Nc = nt * 16 + (ln & 15);          // 0..31 (valid < 20)
                f1[idx] = (K < 20 && Nc < 20) ? W1_sso[K * 20 + Nc] : 0.f;
            } else {
                const int j  = idx - 640;
                const int v  = j & 1;
                const int ln = (j >> 1) & 31;
                const int kk = j >> 6;              // 0..24
                const int K  = 4 * kk + 2 * (ln >> 4) + v;   // 0..99
                const int Nc = ln & 15;                      // valid < 9
                f2[j] = (Nc < 9) ? W2_sso[K * 9 + Nc] : 0.f;
            }
        }
    }
    __syncthreads();

    const int  lane = tid & 31;
    const int  wv   = tid >> 5;
    const int  half = lane >> 4;    // 0: lanes 0-15 (K0,K1) ; 1: lanes 16-31 (K2,K3)
    const int  ml   = lane & 15;    // edge-in-tile / column-in-tile

    const long e_raw = ((long)blockIdx.x * WPB + wv) * 16 + ml;
    const bool live  = e_raw < (long)E;
    const long e     = live ? e_raw : (long)E - 1;

    // ---------------- geometry: rows 4e+1, 4e+2, 4e+3 ----------------
    const float* lp = lig + 12 * e + 3;
    const float* rp = rec + 12 * e + 3;
    float ev[9];
#pragma unroll
    for (int i = 0; i < 9; ++i) ev[i] = lp[i] - rp[i];

    // dist = sqrt(ss); unit = ev/dist  ==>  r = rsq(ss); dist = ss*r; unit = ev*r
    const float ss1 = ev[0]*ev[0] + ev[1]*ev[1] + ev[2]*ev[2] + 1e-12f;
    const float ss2 = ev[3]*ev[3] + ev[4]*ev[4] + ev[5]*ev[5] + 1e-12f;
    const float ss3 = ev[6]*ev[6] + ev[7]*ev[7] + ev[8]*ev[8] + 1e-12f;
    const float r1 = __builtin_amdgcn_rsqf(ss1);
    const float r2 = __builtin_amdgcn_rsqf(ss2);
    const float r3 = __builtin_amdgcn_rsqf(ss3);
    const float d1 = ss1 * r1, d2 = ss2 * r2, d3 = ss3 * r3;
    float V1[3], V2[3], V3[3];
#pragma unroll
    for (int c = 0; c < 3; ++c) { V1[c] = ev[c]*r1; V2[c] = ev[3+c]*r2; V3[c] = ev[6+c]*r3; }

    // Gaussian smears: linspace(0,5,5) step 1.25; linspace(0,5,4) step 5/3.
    // Argument is always in [-~46, 0] -> raw v_exp_f32 is safe (no range fixups needed).
    float s1[5], s2[4], s3[5];
#pragma unroll
    for (int i = 0; i < 5; ++i) {
        float t = d1 - 1.25f * (float)i;       s1[i] = __builtin_amdgcn_exp2f(C5L2 * t * t);
        float u = d3 - 1.25f * (float)i;       s3[i] = __builtin_amdgcn_exp2f(C5L2 * u * u);
    }
#pragma unroll
    for (int j = 0; j < 4; ++j) {
        float t = d2 - (5.f / 3.f) * (float)j; s2[j] = __builtin_amdgcn_exp2f(C4L2 * t * t);
    }

    const float dot12 = (V1[0]*V2[0] + V1[1]*V2[1] + V1[2]*V2[2]) * IS3;
    float cr12[3];
    cr12[0] = (V1[1]*V2[2] - V1[2]*V2[1]) * IS2;
    cr12[1] = (V1[2]*V2[0] - V1[0]*V2[2]) * IS2;
    cr12[2] = (V1[0]*V2[1] - V1[1]*V2[0]) * IS2;

    // ---------------- layer 1 scalar channel: [16x20] x [20x20] via WMMA f32 ----------------
    // A1[K=i*4+j] = s1[i]*s2[j]; per K-step kk the lane supplies K=4kk+2*half+{0,1}
    const float s2a = half ? s2[2] : s2[0];
    const float s2b = half ? s2[3] : s2[1];
    v8f acc0 = {0,0,0,0,0,0,0,0};
    v8f acc1 = {0,0,0,0,0,0,0,0};
#pragma unroll
    for (int kk = 0; kk < 5; ++kk) {
        v2f a; a.x = s1[kk] * s2a; a.y = s1[kk] * s2b;
        v2f b0 = w1b[kk * 32 + lane];
        acc0 = __builtin_amdgcn_wmma_f32_16x16x4_f32(false, a, false, b0, (short)0, acc0, false, false);
        v2f b1 = w1b[(5 + kk) * 32 + lane];
        acc1 = __builtin_amdgcn_wmma_f32_16x16x4_f32(false, a, false, b1, (short)0, acc1, false, false);
    }

    // D layout: lane L, vgpr v -> (M = v+8*(L/16), N = L%16). Transpose through LDS.
    float* scw = sc[wv];
#pragma unroll
    for (int v = 0; v < 8; ++v) {
        scw[(v + 8 * half) * 32 + ml]      = acc0[v];
        scw[(v + 8 * half) * 32 + 16 + ml] = acc1[v];
    }
    float outS[20];
#pragma unroll
    for (int n = 0; n < 20; ++n)
        outS[n] = (scw[ml * 32 + n] + dot12 * W1_vvo[n]) * RS21;

    // ---------------- layer 1 vector channel (VALU) ----------------
    float ov[5][3];
#pragma unroll
    for (int k = 0; k < 5; ++k) {
        float p = 0.f, q = 0.f;
#pragma unroll
        for (int i = 0; i < 5; ++i) p += s1[i] * W1_svv[i * 5 + k];
#pragma unroll
        for (int j = 0; j < 4; ++j) q += s2[j] * W1_vsv[j * 5 + k];
        const float wv5 = W1_vvv[k];
#pragma unroll
        for (int c = 0; c < 3; ++c)
            ov[k][c] = (p * V2[c] + q * V1[c] + cr12[c] * wv5) * RS10;
    }

    float dot2[5], cr2[5][3];
#pragma unroll
    for (int k = 0; k < 5; ++k) {
        dot2[k] = (ov[k][0]*V3[0] + ov[k][1]*V3[1] + ov[k][2]*V3[2]) * IS3;
        cr2[k][0] = (ov[k][1]*V3[2] - ov[k][2]*V3[1]) * IS2;
        cr2[k][1] = (ov[k][2]*V3[0] - ov[k][0]*V3[2]) * IS2;
        cr2[k][2] = (ov[k][0]*V3[1] - ov[k][1]*V3[0]) * IS2;
    }

    // ---------------- layer 2 scalar channel: [16x100] x [100x9] via WMMA f32 ----------------
    // A2[K=i*5+j] = outS[i]*s3[j]; indices fold at compile time after full unroll
    v8f acc2 = {0,0,0,0,0,0,0,0};
#pragma unroll
    for (int kk = 0; kk < 25; ++kk) {
        const int k0 = 4 * kk, k1 = k0 + 1, k2 = k0 + 2, k3 = k0 + 3;
        const float axl = outS[k0 / 5] * s3[k0 % 5];
        const float ayl = outS[k1 / 5] * s3[k1 % 5];
        const float axh = outS[k2 / 5] * s3[k2 % 5];
        const float ayh = outS[k3 / 5] * s3[k3 % 5];
        v2f a; a.x = half ? axh : axl; a.y = half ? ayh : ayl;
        v2f b = w2b[kk * 32 + lane];
        acc2 = __builtin_amdgcn_wmma_f32_16x16x4_f32(false, a, false, b, (short)0, acc2, false, false);
    }

#pragma unroll
    for (int v = 0; v < 8; ++v)
        scw[(v + 8 * half) * 32 + ml] = acc2[v];

    float osr[9];
#pragma unroll
    for (int o = 0; o < 9; ++o) {
        float t = 0.f;
#pragma unroll
        for (int k = 0; k < 5; ++k) t += dot2[k] * W2_vvo[k * 9 + o];
        osr[o] = (scw[ml * 32 + o] + t) * RS105;
    }

    // ---------------- layer 2 vector channel (VALU) ----------------
    float svv = 0.f;
#pragma unroll
    for (int i = 0; i < 20; ++i) svv += outS[i] * W2_svv[i];
    float mk[5];
#pragma unroll
    for (int k = 0; k < 5; ++k) {
        float m = 0.f;
#pragma unroll
        for (int j = 0; j < 5; ++j) m += s3[j] * W2_vsv[k * 5 + j];
        mk[k] = m;
    }
    float ovo[3];
#pragma unroll
    for (int c = 0; c < 3; ++c) {
        float acc = svv * V3[c];
#pragma unroll
        for (int k = 0; k < 5; ++k) acc += ov[k][c] * mk[k];
#pragma unroll
        for (int k = 0; k < 5; ++k) acc += cr2[k][c] * W2_vvv[k];
        ovo[c] = acc * RS50;
    }

    // ---------------- output: (4, E, 3); lower half writes q=0,1 ; upper half q=2,3 ----------------
    if (live) {
        const float a0 = half ? osr[6] : osr[0];
        const float a1 = half ? osr[7] : osr[1];
        const float a2 = half ? osr[8] : osr[2];
        const float b0 = half ? ovo[0] : osr[3];
        const float b1 = half ? ovo[1] : osr[4];
        const float b2 = half ? ovo[2] : osr[5];
        const long  qa = half ? 2 : 0;
        const long  qb = half ? 3 : 1;
        float* oA = out + qa * (3l * E) + 3 * e;
        oA[0] = a0; oA[1] = a1; oA[2] = a2;
        float* oB = out + qb * (3l * E) + 3 * e;
        oB[0] = b0; oB[1] = b1; oB[2] = b2;
    }
}

extern "C" void kernel_launch(void* const* d_in, const int* in_sizes, int n_in,
                              void* d_out, int out_size, void* d_ws, size_t ws_size,
                              hipStream_t stream) {
    const float* lig    = (const float*)d_in[0];
    const float* rec    = (const float*)d_in[1];
    const float* W1_sso = (const float*)d_in[2];
    const float* W1_svv = (const float*)d_in[3];
    const float* W1_vsv = (const float*)d_in[4];
    const float* W1_vvo = (const float*)d_in[5];
    const float* W1_vvv = (const float*)d_in[6];
    const float* W2_sso = (const float*)d_in[7];
    const float* W2_svv = (const float*)d_in[8];
    const float* W2_vsv = (const float*)d_in[9];
    const float* W2_vvo = (const float*)d_in[10];
    const float* W2_vvv = (const float*)d_in[11];
    float* outp = (float*)d_out;

    const int E = in_sizes[0] / 12;                 // N = in_sizes[0]/3, E = N/4
    const int edges_per_block = WPB * 16;           // 64
    const int blocks = (E + edges_per_block - 1) / edges_per_block;   // 15625 for E=1e6

    frame_dock_kernel<<<blocks, BLOCK, 0, stream>>>(
        lig, rec, W1_sso, W1_svv, W1_vsv, W1_vvo, W1_vvv,
        W2_sso, W2_svv, W2_vsv, W2_vvo, W2_vvv, outp, E);
}